// VectorQuantizer_54863912239833
// MI455X (gfx1250) — compile-verified
//
#include <hip/hip_runtime.h>
#include <hip/hip_bf16.h>
#include <stdint.h>

typedef float v2f __attribute__((ext_vector_type(2)));
typedef float v8f __attribute__((ext_vector_type(8)));

#define DECAY_F      0.99f
#define ONE_M_DECAY  0.01f
#define COMMIT_F     0.25f
#define EPS_F        1e-5f
#define DD           64            // embedding dim (reference D)

// ---------------------------------------------------------------------------
// 0) zero scratch accumulators (counts[K], dw[K*D], scalars[2])
// ---------------------------------------------------------------------------
__global__ void vq_zero_kernel(float* __restrict__ counts, float* __restrict__ dw,
                               float* __restrict__ scalars, int K, int KD) {
    int t = blockIdx.x * blockDim.x + threadIdx.x;
    if (t < KD) dw[t] = 0.0f;
    if (t < K)  counts[t] = 0.0f;
    if (t < 2)  scalars[t] = 0.0f;
}

// ---------------------------------------------------------------------------
// 1) wn[k] = -0.5 * ||w_k||^2   (per-column bias applied at argmax time)
// ---------------------------------------------------------------------------
__global__ void vq_wnorm_kernel(const float* __restrict__ w, float* __restrict__ wn, int K) {
    int k = blockIdx.x * blockDim.x + threadIdx.x;
    if (k >= K) return;
    const float4* row = (const float4*)(w + (size_t)k * DD);
    float s = 0.0f;
#pragma unroll
    for (int i = 0; i < DD / 4; ++i) {
        float4 v = row[i];
        s += v.x * v.x + v.y * v.y + v.z * v.z + v.w * v.w;
    }
    wn[k] = -0.5f * s;
}

// ---------------------------------------------------------------------------
// 2) fused GEMM + argmin via V_WMMA_F32_16X16X4_F32 with async-LDS staging.
//    score[m][n] = x_m . w_n - 0.5*||w_n||^2 ;  idx[m] = argmax_n score
//    (== argmin of squared distance; strict-greater update + min-index
//     tie-break reproduces jnp.argmin first-occurrence semantics)
//
//    Block = 4 waves; each wave owns one 16-row A tile (kept in 32 VGPRs).
//    The shared 16-code B tile (4 KB) is double-buffered in LDS and
//    prefetched with GLOBAL_LOAD_ASYNC_TO_LDS_B128 (ASYNCcnt) while the
//    current tile feeds 16 chained WMMAs. sched_group_barrier pins a
//    4-deep DS-read pipeline ahead of the WMMA chain so dscnt waits are
//    staircase instead of 0x0; C starts at inline 0 (wn bias folded into
//    the compare) so no load gates the chain start.
//
//    LDS layout (dwords): Btile[buf][c*64 + 4*row + 2*half + v]
//      - global 16B w[row][4c..4c+3] -> contiguous 16B in LDS (b128 fill ok)
//      - wave read per chunk: lane L -> dword 4*(L&15)+2*(L>>4): all 64 banks
// ---------------------------------------------------------------------------
__global__ __launch_bounds__(128)
void vq_argmin_wmma_kernel(const float* __restrict__ x, const float* __restrict__ w,
                           const float* __restrict__ wn, int* __restrict__ idx_out,
                           float* __restrict__ enc_out, int K) {
    __shared__ __attribute__((aligned(16))) float Btile[2][1024];  // 2 x 4 KB

    const int tid  = threadIdx.x;
    const int lane = tid & 31;
    const int wave = tid >> 5;
    const int m16  = lane & 15;      // row (A) / col (B,C) within tile
    const int half = lane >> 4;      // selects K-pair (A/B) and M-half (C)
    const int r0   = (blockIdx.x * 4 + wave) * 16;

    // ---- A tile: 16x64 fp32 in 16 v2f chunks (16x4 f32 A layout) ----
    const float* ap = x + (size_t)(r0 + m16) * DD + 2 * half;
    v2f A[16];
#pragma unroll
    for (int c = 0; c < 16; ++c) A[c] = *(const v2f*)(ap + 4 * c);

    // ---- cooperative async fill: 256 b128 transfers/tile, 2 per thread ----
    const int e0 = tid, e1 = tid + 128;          // element = (chunk c, row)
    const int c0 = e0 >> 4, rw0 = e0 & 15;
    const int c1 = e1 >> 4, rw1 = e1 & 15;

    auto fill = [&](int colbase, int buf) {
        const float* g0 = w + (size_t)(colbase + rw0) * DD + 4 * c0;
        const float* g1 = w + (size_t)(colbase + rw1) * DD + 4 * c1;
        unsigned l0 = (unsigned)(uintptr_t)&Btile[buf][c0 * 64 + 4 * rw0];
        unsigned l1 = (unsigned)(uintptr_t)&Btile[buf][c1 * 64 + 4 * rw1];
        asm volatile("global_load_async_to_lds_b128 %0, %1, off"
                     :: "v"(l0), "v"(g0) : "memory");
        asm volatile("global_load_async_to_lds_b128 %0, %1, off"
                     :: "v"(l1), "v"(g1) : "memory");
    };

    float best_val[8];
    int   best_idx[8];
#pragma unroll
    for (int r = 0; r < 8; ++r) { best_val[r] = -3.402823466e38f; best_idx[r] = 0; }

    const int ntiles = K / 16;
    fill(0, 0);
    asm volatile("s_wait_asynccnt 0x0" ::: "memory");
    __syncthreads();

    for (int t = 0; t < ntiles; ++t) {
        const int cur     = t & 1;
        const int colbase = t * 16;
        if (t + 1 < ntiles) fill(colbase + 16, cur ^ 1);  // prefetch next tile

        float wnl = wn[colbase + m16];     // off critical path: used after WMMAs

        const float* lb = &Btile[cur][4 * m16 + 2 * half];
        v2f Bv[16];
#pragma unroll
        for (int c = 0; c < 16; ++c) Bv[c] = *(const v2f*)(lb + 64 * c);

        v8f C = {};                        // inline 0 as SRC2 for first WMMA
#pragma unroll
        for (int c = 0; c < 16; ++c) {
            C = __builtin_amdgcn_wmma_f32_16x16x4_f32(
                    /*neg_a=*/false, A[c], /*neg_b=*/false, Bv[c],
                    /*c_mod=*/(short)0, C, /*reuse_a=*/false, /*reuse_b=*/false);
        }

        // Pin the pipeline: 4 DS-reads (8 chunks) in flight, then 2 WMMA per
        // extra DS-read, tail of 8 WMMAs.  0x100 = DS read, 0x8 = MFMA/WMMA.
        __builtin_amdgcn_sched_group_barrier(0x100, 4, 0);
        __builtin_amdgcn_sched_group_barrier(0x008, 2, 0);
        __builtin_amdgcn_sched_group_barrier(0x100, 1, 0);
        __builtin_amdgcn_sched_group_barrier(0x008, 2, 0);
        __builtin_amdgcn_sched_group_barrier(0x100, 1, 0);
        __builtin_amdgcn_sched_group_barrier(0x008, 2, 0);
        __builtin_amdgcn_sched_group_barrier(0x100, 1, 0);
        __builtin_amdgcn_sched_group_barrier(0x008, 2, 0);
        __builtin_amdgcn_sched_group_barrier(0x100, 1, 0);
        __builtin_amdgcn_sched_group_barrier(0x008, 8, 0);

        const int cand = colbase + m16;    // this lane's candidate code index
#pragma unroll
        for (int r = 0; r < 8; ++r) {
            float v = C[r] + wnl;          // row M = r + 8*half, col N = m16
            if (v > best_val[r]) { best_val[r] = v; best_idx[r] = cand; }
        }

        asm volatile("s_wait_asynccnt 0x0" ::: "memory");  // next tile resident
        __syncthreads();                                   // all waves done with cur
    }

    // Reduce across the 16 lanes of each half (cols); min-index on ties.
#pragma unroll
    for (int m = 1; m <= 8; m <<= 1) {
#pragma unroll
        for (int r = 0; r < 8; ++r) {
            float ov = __shfl_xor(best_val[r], m, 32);
            int   oi = __shfl_xor(best_idx[r], m, 32);
            if (ov > best_val[r] || (ov == best_val[r] && oi < best_idx[r])) {
                best_val[r] = ov; best_idx[r] = oi;
            }
        }
    }
    if (m16 == 0) {
#pragma unroll
        for (int r = 0; r < 8; ++r) {
            int row = r0 + r + 8 * half;
            idx_out[row] = best_idx[r];
            enc_out[row] = (float)best_idx[r];
        }
    }
}

// ---------------------------------------------------------------------------
// 3) gather quantized, segment-sum counts/dw via fp32 atomics, SSE for loss
// ---------------------------------------------------------------------------
__global__ void vq_gather_kernel(const float* __restrict__ x, const float* __restrict__ w,
                                 const int* __restrict__ idx, float* __restrict__ quant,
                                 float* __restrict__ dw, float* __restrict__ counts,
                                 float* __restrict__ sse) {
    int t = blockIdx.x * blockDim.x + threadIdx.x;
    int i = t >> 6, d = t & 63;
    int k = idx[i];
    float xv = x[t];
    float qv = w[(size_t)k * DD + d];
    quant[t] = qv;                                  // quantized_st values == quantized
    atomicAdd(dw + (size_t)k * DD + d, xv);         // encodings.T @ flat
    if (d == 0) atomicAdd(counts + k, 1.0f);        // segment counts
    float diff = qv - xv;
    float s = diff * diff;
#pragma unroll
    for (int m = 1; m < 32; m <<= 1) s += __shfl_xor(s, m, 32);
    if ((threadIdx.x & 31) == 0) atomicAdd(sse, s);
}

// ---------------------------------------------------------------------------
// 4a) cs_raw = ema_cs*decay + (1-decay)*counts ; n = sum(cs_raw)
//     (cs_raw overwrites counts in-place in scratch)
// ---------------------------------------------------------------------------
__global__ void vq_ema_a_kernel(const float* __restrict__ ema_cs, float* __restrict__ counts_cs,
                                float* __restrict__ n_acc, int K) {
    int k = blockIdx.x * blockDim.x + threadIdx.x;
    float csr = 0.0f;
    if (k < K) {
        csr = ema_cs[k] * DECAY_F + ONE_M_DECAY * counts_cs[k];
        counts_cs[k] = csr;
    }
    float s = csr;
#pragma unroll
    for (int m = 1; m < 32; m <<= 1) s += __shfl_xor(s, m, 32);
    if ((threadIdx.x & 31) == 0) atomicAdd(n_acc, s);
}

// ---------------------------------------------------------------------------
// 4b) Laplace smoothing, new_ema_w, new_weight, cs, loss
// ---------------------------------------------------------------------------
__global__ void vq_ema_b_kernel(const float* __restrict__ ema_w, const float* __restrict__ dw,
                                const float* __restrict__ cs_raw, const float* __restrict__ n_acc,
                                const float* __restrict__ sse,
                                float* __restrict__ out_new_w, float* __restrict__ out_cs,
                                float* __restrict__ out_new_ema_w, float* __restrict__ out_loss,
                                int K, float inv_ND) {
    int t = blockIdx.x * blockDim.x + threadIdx.x;
    int k = t >> 6, d = t & 63;
    float n  = *n_acc;
    float cs = (cs_raw[k] + EPS_F) / (n + (float)K * EPS_F) * n;
    float ne = ema_w[t] * DECAY_F + ONE_M_DECAY * dw[t];
    out_new_ema_w[t] = ne;
    out_new_w[t]     = ne / cs;
    if (d == 0) out_cs[k] = cs;
    if (t == 0) *out_loss = (1.0f + COMMIT_F) * (*sse) * inv_ND;  // q_latent + 0.25*e_latent
}

// ---------------------------------------------------------------------------
extern "C" void kernel_launch(void* const* d_in, const int* in_sizes, int n_in,
                              void* d_out, int out_size, void* d_ws, size_t ws_size,
                              hipStream_t stream) {
    const float* x       = (const float*)d_in[0];  // [B,T,D]
    const float* weight  = (const float*)d_in[1];  // [K,D]
    const float* ema_cs  = (const float*)d_in[2];  // [K]
    const float* ema_w   = (const float*)d_in[3];  // [K,D]

    const int K  = in_sizes[2];
    const int KD = in_sizes[1];
    const int N  = in_sizes[0] / DD;       // B*T
    const int ND = N * DD;

    // scratch layout (floats): wn[K] | counts/cs_raw[K] | dw[K*D] | idx[N] | sse, n
    float* wsf     = (float*)d_ws;
    float* wn      = wsf;
    float* counts  = wsf + K;
    float* dw      = wsf + 2 * K;
    int*   idxbuf  = (int*)(wsf + 2 * K + KD);
    float* scalars = wsf + 2 * K + KD + N;
    float* sse     = scalars;
    float* nacc    = scalars + 1;

    // d_out layout: quantized_st[N*D] | loss[1] | enc_idx[N] | new_w[K*D] | cs[K] | new_ema_w[K*D]
    float* out         = (float*)d_out;
    float* out_quant   = out;
    float* out_loss    = out + ND;
    float* out_enc     = out_loss + 1;
    float* out_new_w   = out_enc + N;
    float* out_cs      = out_new_w + KD;
    float* out_new_ema = out_cs + K;

    vq_zero_kernel<<<(KD + 255) / 256, 256, 0, stream>>>(counts, dw, scalars, K, KD);
    vq_wnorm_kernel<<<(K + 255) / 256, 256, 0, stream>>>(weight, wn, K);
    vq_argmin_wmma_kernel<<<N / 64, 128, 0, stream>>>(x, weight, wn, idxbuf, out_enc, K);
    vq_gather_kernel<<<ND / 256, 256, 0, stream>>>(x, weight, idxbuf, out_quant, dw, counts, sse);
    vq_ema_a_kernel<<<(K + 255) / 256, 256, 0, stream>>>(ema_cs, counts, nacc, K);
    vq_ema_b_kernel<<<KD / 256, 256, 0, stream>>>(ema_w, dw, counts, nacc, sse,
                                                  out_new_w, out_cs, out_new_ema, out_loss,
                                                  K, 1.0f / (float)ND);
}